// ProteinEncoder_31533649887772
// MI455X (gfx1250) — compile-verified
//
#include <hip/hip_runtime.h>

#define HID 128
#define NEG_SLOPE 0.2f
#define EPS_BN 1e-5f
#define NGRAPH 64

typedef __attribute__((ext_vector_type(2))) float v2f;
typedef __attribute__((ext_vector_type(8))) float v8f;

__device__ __forceinline__ float lrelu(float x) { return x > 0.f ? x : NEG_SLOPE * x; }

// float atomic max via sign-split monotonic int/uint trick (works for mixed signs,
// location must be initialized to -inf)
__device__ __forceinline__ void atomicMaxF(float* addr, float val) {
    if (val >= 0.f) atomicMax((int*)addr, __float_as_int(val));
    else            atomicMin((unsigned int*)addr, __float_as_uint(val));
}

// ---------------- WMMA f32 GEMM: H[N,HID] = X[N,K] @ W[K,HID] ----------------
// one wave per 16x16 output tile; K stepped by 4 (V_WMMA_F32_16X16X4_F32)
__global__ void gemm_wmma_f32(const float* __restrict__ X, const float* __restrict__ W,
                              float* __restrict__ H, int N, int K) {
    int tm = blockIdx.x;           // M tile (16 rows)
    int tn = blockIdx.y;           // N tile (16 cols)
    int l  = threadIdx.x;          // 0..31
    int half = l >> 4;             // K sub-pair select (K=0,1 vs K=2,3)
    int li   = l & 15;
    int m = tm * 16 + li;          // A row for this lane
    int n = tn * 16 + li;          // B col for this lane
    v8f acc = {0.f, 0.f, 0.f, 0.f, 0.f, 0.f, 0.f, 0.f};
    const float* xrow = X + (size_t)m * K;
    for (int kk = 0; kk < K; kk += 4) {
        int k0 = kk + half * 2;
        v2f a; a.x = xrow[k0];            a.y = xrow[k0 + 1];
        v2f b; b.x = W[k0 * HID + n];     b.y = W[(k0 + 1) * HID + n];
        acc = __builtin_amdgcn_wmma_f32_16x16x4_f32(false, a, false, b,
                                                    (short)0, acc, false, false);
    }
#pragma unroll
    for (int v = 0; v < 8; ++v) {
        int row = tm * 16 + half * 8 + v;     // C/D layout: vgpr v -> row v (+8 for lanes 16-31)
        H[(size_t)row * HID + tn * 16 + li] = acc[v];
    }
}

// a_s[i] = h[i]·att_src ; a_d[i] = h[i]·att_dst   (one wave per node)
__global__ void node_dots(const float* __restrict__ H, const float* __restrict__ att_s,
                          const float* __restrict__ att_d, float* __restrict__ a_s,
                          float* __restrict__ a_d, int N) {
    int wave = (int)((blockIdx.x * (size_t)blockDim.x + threadIdx.x) >> 5);
    int lane = threadIdx.x & 31;
    if (wave >= N) return;
    float4 h4 = ((const float4*)(H + (size_t)wave * HID))[lane];
    float4 sa = ((const float4*)att_s)[lane];
    float4 da = ((const float4*)att_d)[lane];
    float vs = h4.x * sa.x + h4.y * sa.y + h4.z * sa.z + h4.w * sa.w;
    float vd = h4.x * da.x + h4.y * da.y + h4.z * da.z + h4.w * da.w;
    for (int off = 16; off; off >>= 1) {
        vs += __shfl_xor(vs, off, 32);
        vd += __shfl_xor(vd, off, 32);
    }
    if (lane == 0) { a_s[wave] = vs; a_d[wave] = vd; }
}

// watt[j] = sum_c We[j,c] * att_edge[c]   (EDGE_DIM = 3)
__global__ void compute_watt(const float* __restrict__ We, const float* __restrict__ att_e,
                             float* __restrict__ watt) {
    int j = threadIdx.x;
    if (j < 3) {
        float s = 0.f;
        for (int c = 0; c < HID; ++c) s += We[j * HID + c] * att_e[c];
        watt[j] = s;
    }
}

__global__ void init_layer(float* __restrict__ out, const float* __restrict__ bias,
                           float* deg, float* sumae, float* amax, float* denom,
                           float* bnsum, float* bnsq, int N) {
    size_t i = (size_t)blockIdx.x * blockDim.x + threadIdx.x;
    if (i < (size_t)N * HID) out[i] = bias[i & (HID - 1)];
    if (i < (size_t)N) {
        deg[i] = 0.f; sumae[i] = 0.f; denom[i] = 0.f;
        amax[i] = __uint_as_float(0xff800000u);   // -inf
    }
    if (i < HID) { bnsum[i] = 0.f; bnsq[i] = 0.f; }
}

// per-edge a_e, degree, and sum of a_e per destination (for self-loop mean attr)
__global__ void edge_pre(const int* __restrict__ ei, const float* __restrict__ eattr,
                         const float* __restrict__ watt, float* __restrict__ ae_buf,
                         float* __restrict__ deg, float* __restrict__ sumae, int E) {
    int e = blockIdx.x * blockDim.x + threadIdx.x;
    if (e >= E) return;
    int d = ei[E + e];
    float ae = eattr[3 * e] * watt[0] + eattr[3 * e + 1] * watt[1] + eattr[3 * e + 2] * watt[2];
    ae_buf[e] = ae;
    atomicAdd(&deg[d], 1.f);
    atomicAdd(&sumae[d], ae);
}

__global__ void edge_alpha_max(const int* __restrict__ ei, const float* __restrict__ a_s,
                               const float* __restrict__ a_d, float* __restrict__ ae_buf,
                               float* __restrict__ amax, int E) {
    int e = blockIdx.x * blockDim.x + threadIdx.x;
    if (e >= E) return;
    int s = ei[e], d = ei[E + e];
    float al = lrelu(a_s[s] + a_d[d] + ae_buf[e]);
    ae_buf[e] = al;
    atomicMaxF(&amax[d], al);
}

// self-loop alpha (edge attr = mean of incoming) + fold into segment max
__global__ void node_loop_alpha(const float* __restrict__ a_s, const float* __restrict__ a_d,
                                const float* __restrict__ deg, const float* __restrict__ sumae,
                                float* __restrict__ amax, float* __restrict__ aloop, int N) {
    int i = blockIdx.x * blockDim.x + threadIdx.x;
    if (i >= N) return;
    float lae = sumae[i] / fmaxf(deg[i], 1.f);
    float al = lrelu(a_s[i] + a_d[i] + lae);
    aloop[i] = al;
    amax[i] = fmaxf(amax[i], al);
}

__global__ void edge_expsum(const int* __restrict__ ei, float* __restrict__ ae_buf,
                            const float* __restrict__ amax, float* __restrict__ denom, int E) {
    int e = blockIdx.x * blockDim.x + threadIdx.x;
    if (e >= E) return;
    int d = ei[E + e];
    float ex = expf(ae_buf[e] - amax[d]);
    ae_buf[e] = ex;
    atomicAdd(&denom[d], ex);
}

__global__ void node_loop_exp(float* __restrict__ aloop, const float* __restrict__ amax,
                              float* __restrict__ denom, int N) {
    int i = blockIdx.x * blockDim.x + threadIdx.x;
    if (i >= N) return;
    float ex = expf(aloop[i] - amax[i]);
    aloop[i] = ex;
    denom[i] += ex;   // safe: runs after edge_expsum completed (stream order)
}

// heavy pass: out[dst] += w * h[src]; one lane handles 4 channels (float4 gather)
__global__ void edge_scatter(const int* __restrict__ ei, const float* __restrict__ ae_buf,
                             const float* __restrict__ denom, const float* __restrict__ H,
                             float* __restrict__ out, int E) {
    long long t = (long long)blockIdx.x * blockDim.x + threadIdx.x;
    int e = (int)(t >> 5);
    int q = (int)(t & 31);
    if (e >= E) return;
    int s = ei[e], d = ei[E + e];
    float w = ae_buf[e] / (denom[d] + 1e-16f);
    float4 hv = ((const float4*)(H + (size_t)s * HID))[q];
    float* o = out + (size_t)d * HID + q * 4;
    atomicAdd(o + 0, w * hv.x);
    atomicAdd(o + 1, w * hv.y);
    atomicAdd(o + 2, w * hv.z);
    atomicAdd(o + 3, w * hv.w);
}

__global__ void node_self_add(const float* __restrict__ aloop, const float* __restrict__ denom,
                              const float* __restrict__ H, float* __restrict__ out, int N) {
    long long t = (long long)blockIdx.x * blockDim.x + threadIdx.x;
    int i = (int)(t >> 5);
    int q = (int)(t & 31);
    if (i >= N) return;
    float w = aloop[i] / (denom[i] + 1e-16f);
    float4 hv = ((const float4*)(H + (size_t)i * HID))[q];
    float4* o = (float4*)(out + (size_t)i * HID) + q;
    float4 c = *o;
    c.x += w * hv.x; c.y += w * hv.y; c.z += w * hv.z; c.w += w * hv.w;
    *o = c;
}

__global__ void bn_stats(const float* __restrict__ H, float* __restrict__ sum,
                         float* __restrict__ sumsq, int N) {
    int c = threadIdx.x;                       // 128 channels
    float s = 0.f, q = 0.f;
    for (int r = blockIdx.x; r < N; r += gridDim.x) {
        float v = H[(size_t)r * HID + c];
        s += v; q += v * v;
    }
    atomicAdd(&sum[c], s);
    atomicAdd(&sumsq[c], q);
}

__global__ void bn_apply_relu(float* __restrict__ H, const float* __restrict__ sum,
                              const float* __restrict__ sumsq, const float* __restrict__ gamma,
                              const float* __restrict__ beta, int N) {
    size_t t = (size_t)blockIdx.x * blockDim.x + threadIdx.x;
    if (t >= (size_t)N * HID) return;
    int c = (int)(t & (HID - 1));
    float invN = 1.f / (float)N;
    float mu = sum[c] * invN;
    float var = sumsq[c] * invN - mu * mu;
    float v = gamma[c] * (H[t] - mu) * rsqrtf(var + EPS_BN) + beta[c];
    H[t] = fmaxf(v, 0.f);
}

__global__ void pool_zero(float* __restrict__ gsum, float* __restrict__ gcnt) {
    int t = blockIdx.x * blockDim.x + threadIdx.x;
    if (t < NGRAPH * HID) gsum[t] = 0.f;
    if (t < NGRAPH) gcnt[t] = 0.f;
}

__global__ void pool_acc(const float* __restrict__ H, const int* __restrict__ batch,
                         float* __restrict__ gsum, float* __restrict__ gcnt, int N) {
    long long t = (long long)blockIdx.x * blockDim.x + threadIdx.x;
    int i = (int)(t >> 5);
    int q = (int)(t & 31);
    if (i >= N) return;
    int g = batch[i];
    float4 hv = ((const float4*)(H + (size_t)i * HID))[q];
    float* o = gsum + g * HID + q * 4;
    atomicAdd(o + 0, hv.x);
    atomicAdd(o + 1, hv.y);
    atomicAdd(o + 2, hv.z);
    atomicAdd(o + 3, hv.w);
    if (q == 0) atomicAdd(&gcnt[g], 1.f);
}

__global__ void pool_final(const float* __restrict__ gsum, const float* __restrict__ gcnt,
                           float* __restrict__ out) {
    int t = blockIdx.x * blockDim.x + threadIdx.x;
    if (t < NGRAPH * HID) out[t] = gsum[t] / fmaxf(gcnt[t >> 7], 1.f);
}

static void run_layer(const float* Xin, int K, const float* W, const float* atts,
                      const float* attd, const float* We, const float* atte,
                      const float* bias, const float* gamma, const float* beta,
                      const int* ei, const float* eattr, int N, int E,
                      float* H, float* out, float* ae_buf, float* a_s, float* a_d,
                      float* deg, float* sumae, float* amax, float* denom, float* aloop,
                      float* watt, float* bnsum, float* bnsq, hipStream_t stream) {
    dim3 gg(N / 16, HID / 16);
    gemm_wmma_f32<<<gg, 32, 0, stream>>>(Xin, W, H, N, K);
    node_dots<<<(unsigned)((N * 32 + 127) / 128), 128, 0, stream>>>(H, atts, attd, a_s, a_d, N);
    compute_watt<<<1, 32, 0, stream>>>(We, atte, watt);
    size_t total = (size_t)N * HID;
    init_layer<<<(unsigned)((total + 255) / 256), 256, 0, stream>>>(out, bias, deg, sumae, amax,
                                                                    denom, bnsum, bnsq, N);
    edge_pre<<<(unsigned)((E + 255) / 256), 256, 0, stream>>>(ei, eattr, watt, ae_buf, deg, sumae, E);
    edge_alpha_max<<<(unsigned)((E + 255) / 256), 256, 0, stream>>>(ei, a_s, a_d, ae_buf, amax, E);
    node_loop_alpha<<<(unsigned)((N + 255) / 256), 256, 0, stream>>>(a_s, a_d, deg, sumae, amax, aloop, N);
    edge_expsum<<<(unsigned)((E + 255) / 256), 256, 0, stream>>>(ei, ae_buf, amax, denom, E);
    node_loop_exp<<<(unsigned)((N + 255) / 256), 256, 0, stream>>>(aloop, amax, denom, N);
    long long sc = (long long)E * 32;
    edge_scatter<<<(unsigned)((sc + 255) / 256), 256, 0, stream>>>(ei, ae_buf, denom, H, out, E);
    node_self_add<<<(unsigned)((N * 32LL + 255) / 256), 256, 0, stream>>>(aloop, denom, H, out, N);
    bn_stats<<<256, HID, 0, stream>>>(out, bnsum, bnsq, N);
    bn_apply_relu<<<(unsigned)((total + 255) / 256), 256, 0, stream>>>(out, bnsum, bnsq, gamma, beta, N);
}

extern "C" void kernel_launch(void* const* d_in, const int* in_sizes, int n_in,
                              void* d_out, int out_size, void* d_ws, size_t ws_size,
                              hipStream_t stream) {
    const float* x     = (const float*)d_in[0];
    const int*   ei    = (const int*)d_in[1];
    const float* eattr = (const float*)d_in[2];
    const int*   batch = (const int*)d_in[3];
    const float* W1    = (const float*)d_in[4];
    const float* atts1 = (const float*)d_in[5];
    const float* attd1 = (const float*)d_in[6];
    const float* We1   = (const float*)d_in[7];
    const float* atte1 = (const float*)d_in[8];
    const float* b1    = (const float*)d_in[9];
    const float* W2    = (const float*)d_in[10];
    const float* atts2 = (const float*)d_in[11];
    const float* attd2 = (const float*)d_in[12];
    const float* We2   = (const float*)d_in[13];
    const float* atte2 = (const float*)d_in[14];
    const float* b2    = (const float*)d_in[15];
    const float* g1    = (const float*)d_in[16];
    const float* be1   = (const float*)d_in[17];
    const float* g2    = (const float*)d_in[18];
    const float* be2   = (const float*)d_in[19];

    const int IN = 64;
    int N = in_sizes[0] / IN;     // 50000
    int E = in_sizes[1] / 2;      // 1600000

    float* ws = (float*)d_ws;
    float* B0     = ws; ws += (size_t)N * HID;
    float* B1     = ws; ws += (size_t)N * HID;
    float* ae_buf = ws; ws += E;
    float* a_s    = ws; ws += N;
    float* a_d    = ws; ws += N;
    float* deg    = ws; ws += N;
    float* sumae  = ws; ws += N;
    float* amax   = ws; ws += N;
    float* denom  = ws; ws += N;
    float* aloop  = ws; ws += N;
    float* watt   = ws; ws += 4;
    float* bnsum  = ws; ws += HID;
    float* bnsq   = ws; ws += HID;
    float* gsum   = ws; ws += NGRAPH * HID;
    float* gcnt   = ws; ws += NGRAPH;

    // layer 1: x[N,64] -> B1[N,128] (BN+ReLU in place); h1 in B0
    run_layer(x, IN, W1, atts1, attd1, We1, atte1, b1, g1, be1, ei, eattr, N, E,
              B0, B1, ae_buf, a_s, a_d, deg, sumae, amax, denom, aloop,
              watt, bnsum, bnsq, stream);
    // layer 2: B1[N,128] -> B1 again (gemm consumes B1 into B0 before B1 is reused as out)
    run_layer(B1, HID, W2, atts2, attd2, We2, atte2, b2, g2, be2, ei, eattr, N, E,
              B0, B1, ae_buf, a_s, a_d, deg, sumae, amax, denom, aloop,
              watt, bnsum, bnsq, stream);

    pool_zero<<<(NGRAPH * HID + 255) / 256, 256, 0, stream>>>(gsum, gcnt);
    pool_acc<<<(unsigned)(((long long)N * 32 + 255) / 256), 256, 0, stream>>>(B1, batch, gsum, gcnt, N);
    pool_final<<<(NGRAPH * HID + 255) / 256, 256, 0, stream>>>(gsum, gcnt, (float*)d_out);
}